// IsoMaxLossFirstPartV1_24696061952726
// MI455X (gfx1250) — compile-verified
//
#include <hip/hip_runtime.h>
#include <math.h>

// IsoMaxLossFirstPart for MI455X (gfx1250), fp32 WMMA path.
//
// out[i,j] = -( sqrt(2 + D*eps^2 - 2*<f_i,w_j> + 2*eps*(sf_i - sw_j)) - 0.1*S ) * scale
// where f,w are L2-normalized rows, sf/sw their row sums, and
// S = sum over all (i,j) of the same expression on (w_i, w_j).
//
// GEMMs run on V_WMMA_F32_16X16X4_F32 (wave32, A/B = 2 VGPRs f32, C/D = 8 VGPRs f32).

typedef __attribute__((ext_vector_type(2))) float v2f;
typedef __attribute__((ext_vector_type(8))) float v8f;

#define NB 1024
#define ND 512
#define NC 256
#define EPS_PD 1e-6f
#define EPS_NORM 1e-12f

// ---------------------------------------------------------------------------
// Row L2-normalization (512 cols), also emits row sum of the normalized row.
// One block (128 threads = 4 waves) per row; fixed-shape shuffle+LDS reduction
// (deterministic order).
// ---------------------------------------------------------------------------
__global__ __launch_bounds__(128) void normalize_rows_512(
    const float* __restrict__ x, float* __restrict__ xn,
    float* __restrict__ rowsum)
{
  const int row = blockIdx.x;
  const int tid = threadIdx.x;
  const float* xr = x + (size_t)row * ND;

  float v[4];
  float s = 0.f, q = 0.f;
#pragma unroll
  for (int i = 0; i < 4; ++i) {
    v[i] = xr[tid + i * 128];
    s += v[i];
    q += v[i] * v[i];
  }
  // wave32 reduction
#pragma unroll
  for (int off = 16; off > 0; off >>= 1) {
    s += __shfl_down(s, off, 32);
    q += __shfl_down(q, off, 32);
  }
  __shared__ float rs[4], rq[4];
  __shared__ float sh_inv;
  const int wv = tid >> 5;
  if ((tid & 31) == 0) { rs[wv] = s; rq[wv] = q; }
  __syncthreads();
  if (tid == 0) {
    const float ts = rs[0] + rs[1] + rs[2] + rs[3];
    const float tq = rq[0] + rq[1] + rq[2] + rq[3];
    const float den = fmaxf(sqrtf(tq), EPS_NORM);
    const float inv = 1.0f / den;
    sh_inv = inv;
    rowsum[row] = ts * inv;   // sum of normalized row
  }
  __syncthreads();
  const float inv = sh_inv;
  float* xo = xn + (size_t)row * ND;
#pragma unroll
  for (int i = 0; i < 4; ++i) xo[tid + i * 128] = v[i] * inv;
}

// ---------------------------------------------------------------------------
// Prototype-prototype term: P = wN * wN^T (256x256) via f32 WMMA, fused into
// per-wave partial sums of sqrt(2 + D*eps^2 - 2*P[i,j] + 2*eps*(sw_i - sw_j)).
// 4 waves / block, 1 tile (16x16) per wave, 64 blocks -> 256 tiles.
// ---------------------------------------------------------------------------
__global__ __launch_bounds__(128) void protoc_kernel(
    const float* __restrict__ wN, const float* __restrict__ sw,
    float* __restrict__ partials)
{
  const int lane = threadIdx.x & 31;
  const int wv   = threadIdx.x >> 5;
  const int tile = blockIdx.x * 4 + wv;          // 0..255
  const int rowBase = (tile >> 4) * 16;
  const int colBase = (tile & 15) * 16;
  const int lidx  = lane & 15;                   // A-row / B-col within tile
  const int khalf = (lane >> 4) * 2;             // K pair selected by half-wave

  const float* aPtr = wN + (size_t)(rowBase + lidx) * ND + khalf;
  const float* bPtr = wN + (size_t)(colBase + lidx) * ND + khalf;

  v8f acc = {};
#pragma unroll 4
  for (int k = 0; k < ND; k += 4) {
    v2f a = *(const v2f*)(aPtr + k);
    v2f b = *(const v2f*)(bPtr + k);
    acc = __builtin_amdgcn_wmma_f32_16x16x4_f32(
        /*neg_a=*/false, a, /*neg_b=*/false, b,
        /*c_mod=*/(short)0, acc, /*reuse_a=*/false, /*reuse_b=*/false);
  }

  const float cst  = 2.0f + (float)ND * EPS_PD * EPS_PD;
  const int   mAdd = (lane >> 4) * 8;            // lanes 16..31 hold M = r+8
  const float swn  = sw[colBase + lidx];
  float local = 0.f;
#pragma unroll
  for (int r = 0; r < 8; ++r) {
    const float d2 = cst - 2.0f * acc[r]
                   + 2.0f * EPS_PD * (sw[rowBase + r + mAdd] - swn);
    local += sqrtf(fmaxf(d2, 0.0f));
  }
#pragma unroll
  for (int off = 16; off > 0; off >>= 1) local += __shfl_down(local, off, 32);
  if (lane == 0) partials[blockIdx.x * 4 + wv] = local;
}

// Deterministic final reduction of the 256 per-wave partials.
__global__ void reduce_partials(const float* __restrict__ partials,
                                float* __restrict__ S)
{
  if (threadIdx.x == 0 && blockIdx.x == 0) {
    float s = 0.f;
    for (int i = 0; i < 256; ++i) s += partials[i];
    S[0] = s;
  }
}

// ---------------------------------------------------------------------------
// Main GEMM G = fN * wN^T (1024x256) + fused distance/affine epilogue.
// Each wave owns a 16x64 output strip: A (float2) loaded once per K-step and
// reused across 4 WMMA accumulators. Block = 4 waves (64 rows x 64 cols).
// Grid = (1024/64, 256/64) = (16, 4).
// ---------------------------------------------------------------------------
__global__ __launch_bounds__(128) void gemm_dist_kernel(
    const float* __restrict__ fN, const float* __restrict__ wN,
    const float* __restrict__ sf, const float* __restrict__ sw,
    const float* __restrict__ Ssum, const float* __restrict__ dscale,
    const int* __restrict__ use_scale, float* __restrict__ out)
{
  const int lane = threadIdx.x & 31;
  const int wv   = threadIdx.x >> 5;
  const int rowBase = (blockIdx.x * 4 + wv) * 16;   // 0..1008
  const int colBase = blockIdx.y * 64;              // 0..192
  const int lidx  = lane & 15;
  const int khalf = (lane >> 4) * 2;

  const float* aPtr = fN + (size_t)(rowBase + lidx) * ND + khalf;
  const float* bPtr = wN + (size_t)(colBase + lidx) * ND + khalf;

  v8f acc0 = {}, acc1 = {}, acc2 = {}, acc3 = {};
#pragma unroll 4
  for (int k = 0; k < ND; k += 4) {
    v2f a  = *(const v2f*)(aPtr + k);
    v2f b0 = *(const v2f*)(bPtr + k);
    v2f b1 = *(const v2f*)(bPtr + 16 * ND + k);
    v2f b2 = *(const v2f*)(bPtr + 32 * ND + k);
    v2f b3 = *(const v2f*)(bPtr + 48 * ND + k);
    acc0 = __builtin_amdgcn_wmma_f32_16x16x4_f32(false, a, false, b0, (short)0, acc0, false, false);
    acc1 = __builtin_amdgcn_wmma_f32_16x16x4_f32(false, a, false, b1, (short)0, acc1, false, false);
    acc2 = __builtin_amdgcn_wmma_f32_16x16x4_f32(false, a, false, b2, (short)0, acc2, false, false);
    acc3 = __builtin_amdgcn_wmma_f32_16x16x4_f32(false, a, false, b3, (short)0, acc3, false, false);
  }

  // Epilogue: distance + logits + scaling, fully fused.
  const float S    = Ssum[0];
  const float scl  = use_scale[0] ? fabsf(dscale[0]) : 1.0f;
  const float bias = -0.1f * S;
  const float cst  = 2.0f + (float)ND * EPS_PD * EPS_PD;
  const int   mAdd = (lane >> 4) * 8;

  float swv[4];
#pragma unroll
  for (int t = 0; t < 4; ++t) swv[t] = sw[colBase + t * 16 + lidx];

  v8f accs[4] = {acc0, acc1, acc2, acc3};
#pragma unroll
  for (int r = 0; r < 8; ++r) {
    const int   row = rowBase + r + mAdd;
    const float sfr = sf[row];
    float* orow = out + (size_t)row * NC + colBase + lidx;
#pragma unroll
    for (int t = 0; t < 4; ++t) {
      const float g    = accs[t][r];
      const float d2   = cst - 2.0f * g + 2.0f * EPS_PD * (sfr - swv[t]);
      const float dist = sqrtf(fmaxf(d2, 0.0f));
      orow[t * 16] = -(dist + bias) * scl;
    }
  }
}

// ---------------------------------------------------------------------------
// Host-side launcher (graph-capture safe: only kernel launches on `stream`).
// ---------------------------------------------------------------------------
extern "C" void kernel_launch(void* const* d_in, const int* in_sizes, int n_in,
                              void* d_out, int out_size, void* d_ws, size_t ws_size,
                              hipStream_t stream)
{
  const float* features = (const float*)d_in[0];   // [1024, 512]
  const float* weights  = (const float*)d_in[1];   // [256, 512]
  const float* dscale   = (const float*)d_in[2];   // [1]
  const int*   use_ds   = (const int*)d_in[3];     // [1]
  float* out = (float*)d_out;                      // [1024, 256]

  // Workspace layout (floats): ~2.63 MB total.
  float* ws       = (float*)d_ws;
  float* fN       = ws;                 // 1024*512
  float* wN       = fN + NB * ND;       // 256*512
  float* sf       = wN + NC * ND;       // 1024
  float* sw       = sf + NB;            // 256
  float* partials = sw + NC;            // 256
  float* S        = partials + 256;     // 1

  normalize_rows_512<<<NB, 128, 0, stream>>>(features, fN, sf);
  normalize_rows_512<<<NC, 128, 0, stream>>>(weights, wN, sw);
  protoc_kernel<<<64, 128, 0, stream>>>(wN, sw, partials);
  reduce_partials<<<1, 1, 0, stream>>>(partials, S);
  dim3 grid(16, 4);
  gemm_dist_kernel<<<grid, 128, 0, stream>>>(fN, wN, sf, sw, S, dscale, use_ds, out);
}